// CrossFrameAttention_37331855737420
// MI455X (gfx1250) — compile-verified
//
#include <hip/hip_runtime.h>
#include <math.h>

// CDNA5 / gfx1250, wave32.
// Stage 1: qkv = X @ W^T + b  via V_WMMA_F32_16X16X4_F32 (M=65536,K=512,N=1536),
//          double-buffered LDS filled with GLOBAL_LOAD_ASYNC_TO_LDS_B128 (ASYNCcnt).
// Stage 2: elementwise "cross-frame attention" softmax epilogue (bandwidth bound),
//          K tile staged with GLOBAL_LOAD_ASYNC_TO_LDS_B32.

typedef __attribute__((ext_vector_type(2))) float v2f;
typedef __attribute__((ext_vector_type(4))) float v4f;
typedef __attribute__((ext_vector_type(8))) float v8f;

#define LDA 36   // LDS row stride (floats): 144B = 16B aligned, conflict-free b64 reads
#define LDK 65   // K-tile LDS stride: breaks column-read bank conflicts

// Async global->LDS copies (CDNA5 VGLOBAL async ops, tracked by ASYNCcnt).
// LDS operand is the 32-bit LDS byte address = low 32 bits of the generic pointer
__device__ __forceinline__ void async_copy_b128(const float* g, float* l) {
  asm volatile("global_load_async_to_lds_b128 %0, %1, off"
               :: "v"((unsigned)(size_t)l), "v"((unsigned long long)(size_t)g)
               : "memory");
}
__device__ __forceinline__ void async_copy_b32(const float* g, float* l) {
  asm volatile("global_load_async_to_lds_b32 %0, %1, off"
               :: "v"((unsigned)(size_t)l), "v"((unsigned long long)(size_t)g)
               : "memory");
}
__device__ __forceinline__ void wait_async0() {
  asm volatile("s_wait_asynccnt 0x0" ::: "memory");
}

// ---------------------------------------------------------------------------
// Stage 1: QKV GEMM.  Block tile 128(M) x 64(N), 8 waves (4x2), wave tile 32x32.
// Double-buffered K slabs of 32, async-prefetched one slab ahead.
// ---------------------------------------------------------------------------
__global__ __launch_bounds__(256)
void qkv_gemm_kernel(const float* __restrict__ X, const float* __restrict__ W,
                     const float* __restrict__ bias, float* __restrict__ qkv) {
  __shared__ float sA[2][128 * LDA];
  __shared__ float sB[2][64 * LDA];

  const int tid  = threadIdx.x;
  const int lane = tid & 31;
  const int wid  = tid >> 5;
  const int wm   = wid & 3;        // 4 wave rows  * 32 = 128
  const int wn   = wid >> 2;       // 2 wave cols  * 32 = 64
  const int lrow = lane & 15;      // M (A) / N (B) within 16
  const int lhi  = lane >> 4;      // selects K pair {0,1} vs {2,3}
  const int m0   = blockIdx.x * 128;
  const int n0   = blockIdx.y * 64;

  v8f acc00 = {}, acc01 = {}, acc10 = {}, acc11 = {};

  // issue async copies of one 128x32 A slab + 64x32 B slab into buffer `buf`
  auto stage = [&](int kt, int buf) {
#pragma unroll
    for (int i = 0; i < 4; ++i) {
      int li = tid + i * 256;            // 1024 float4 total (A)
      int r  = li >> 3;
      int c4 = (li & 7) << 2;
      async_copy_b128(X + (size_t)(m0 + r) * 512 + kt + c4, &sA[buf][r * LDA + c4]);
    }
#pragma unroll
    for (int i = 0; i < 2; ++i) {
      int li = tid + i * 256;            // 512 float4 total (B)
      int r  = li >> 3;
      int c4 = (li & 7) << 2;
      async_copy_b128(W + (size_t)(n0 + r) * 512 + kt + c4, &sB[buf][r * LDA + c4]);
    }
  };

  stage(0, 0);
  wait_async0();
  __syncthreads();

  for (int kt = 0; kt < 512; kt += 32) {
    const int buf = (kt >> 5) & 1;
    if (kt + 32 < 512) stage(kt + 32, buf ^ 1);  // overlap next slab with compute

    const float* pA = &sA[buf][0];
    const float* pB = &sB[buf][0];
    // 8 x (K=4) WMMA steps. ISA 16x4 A layout: lane = m + 16*(k>>1), vgpr = k&1.
#pragma unroll
    for (int kk = 0; kk < 32; kk += 4) {
      const int ko = kk + lhi * 2;
      v2f a0 = *(const v2f*)(pA + (wm * 32 +      lrow) * LDA + ko);
      v2f a1 = *(const v2f*)(pA + (wm * 32 + 16 + lrow) * LDA + ko);
      v2f b0 = *(const v2f*)(pB + (wn * 32 +      lrow) * LDA + ko);
      v2f b1 = *(const v2f*)(pB + (wn * 32 + 16 + lrow) * LDA + ko);
      acc00 = __builtin_amdgcn_wmma_f32_16x16x4_f32(false, a0, false, b0, (short)0, acc00, false, false);
      acc01 = __builtin_amdgcn_wmma_f32_16x16x4_f32(false, a0, false, b1, (short)0, acc01, false, false);
      acc10 = __builtin_amdgcn_wmma_f32_16x16x4_f32(false, a1, false, b0, (short)0, acc10, false, false);
      acc11 = __builtin_amdgcn_wmma_f32_16x16x4_f32(false, a1, false, b1, (short)0, acc11, false, false);
    }

    wait_async0();     // drain this wave's prefetch of the next slab
    __syncthreads();   // publish LDS writes / retire all reads of current buf
  }

  // Epilogue: C layout vgpr v -> M = v + 8*(lane>>4), N = lane&15. Fuse bias.
  const int col0  = n0 + wn * 32 + lrow;
  const float bi0 = bias[col0];
  const float bi1 = bias[col0 + 16];
#pragma unroll
  for (int mi = 0; mi < 2; ++mi) {
    v8f aN0 = mi ? acc10 : acc00;
    v8f aN1 = mi ? acc11 : acc01;
#pragma unroll
    for (int v = 0; v < 8; ++v) {
      size_t row = (size_t)(m0 + wm * 32 + mi * 16 + v + 8 * lhi);
      qkv[row * 1536 + col0]      = aN0[v] + bi0;
      qkv[row * 1536 + col0 + 16] = aN1[v] + bi1;
    }
  }
}

// ---------------------------------------------------------------------------
// Stage 2: per (b,n2,fr,h): logit[w][c] = s*Q[tq, fr*64+c]*K[base+c*64+h, 512+frp*64+w]
// softmax over c (64), scale V, scatter to permuted output. 8192 blocks x 256 thr.
// ---------------------------------------------------------------------------
__global__ __launch_bounds__(256)
void attn_softmax_kernel(const float* __restrict__ qkv, float* __restrict__ out) {
  __shared__ float sK[64 * LDK];

  const int tid = threadIdx.x;
  const int bid = blockIdx.x;
  const int h   = bid & 63;
  const int fr  = (bid >> 6) & 7;
  const int n2  = (bid >> 9) & 7;
  const int b   = bid >> 12;
  const int frp = (fr + 7) & 7;
  const size_t base = (size_t)(b * 8 + n2) * 4096;

  // K tile (the one scattered access): sK[c][w] = qkv[(base+c*64+h)*1536 + 512 + frp*64 + w]
  // async global->LDS, coalesced along w.
#pragma unroll
  for (int i = 0; i < 16; ++i) {
    int li = tid + i * 256;
    int c  = li >> 6;
    int w  = li & 63;
    async_copy_b32(qkv + (base + (size_t)c * 64 + h) * 1536 + 512 + (size_t)frp * 64 + w,
                   &sK[c * LDK + w]);
  }
  wait_async0();
  __syncthreads();

  const int w  = tid >> 2;          // 64 spatial-w values
  const int q4 = tid & 3;           // 4 adjacent lanes cooperate on one softmax
  const int c0 = q4 * 16;
  const float* qrow = qkv + (base + (size_t)h * 64 + w) * 1536;
  const float scale = 0.125f;       // 64^-0.5

  float lg[16];
  float mx = -3.402823466e+38f;
#pragma unroll
  for (int j = 0; j < 4; ++j) {
    v4f qv = *(const v4f*)(qrow + fr * 64 + c0 + j * 4);
#pragma unroll
    for (int e = 0; e < 4; ++e) {
      int c = c0 + j * 4 + e;
      float l = scale * qv[e] * sK[c * LDK + w];
      lg[j * 4 + e] = l;
      mx = fmaxf(mx, l);
    }
  }
  mx = fmaxf(mx, __shfl_xor(mx, 1, 4));
  mx = fmaxf(mx, __shfl_xor(mx, 2, 4));

  float s = 0.0f;
#pragma unroll
  for (int i = 0; i < 16; ++i) {
    lg[i] = __expf(lg[i] - mx);
    s += lg[i];
  }
  s += __shfl_xor(s, 1, 4);
  s += __shfl_xor(s, 2, 4);
  const float inv = 1.0f / s;

  // out[b*8 + h/8][(h%8)*8 + w/8][(w%8)*8 + fr][n2*64 + c]
  const int out_b = b * 8 + (h >> 3);
  const int out_h = (h & 7) * 8 + (w >> 3);
  const int out_w = (w & 7) * 8 + fr;
  float* orow = out + (((size_t)out_b * 64 + out_h) * 64 + out_w) * 512 + n2 * 64;
#pragma unroll
  for (int j = 0; j < 4; ++j) {
    v4f vv = *(const v4f*)(qrow + 1024 + frp * 64 + c0 + j * 4);
    v4f o;
#pragma unroll
    for (int e = 0; e < 4; ++e) o[e] = lg[j * 4 + e] * inv * vv[e];
    *(v4f*)(orow + c0 + j * 4) = o;
  }
}

// ---------------------------------------------------------------------------
extern "C" void kernel_launch(void* const* d_in, const int* in_sizes, int n_in,
                              void* d_out, int out_size, void* d_ws, size_t ws_size,
                              hipStream_t stream) {
  const float* X    = (const float*)d_in[0];  // [16,64,64,512]
  const float* W    = (const float*)d_in[1];  // [1536,512]
  const float* bias = (const float*)d_in[2];  // [1536]
  float* out = (float*)d_out;                 // [16,64,64,512]
  float* qkv = (float*)d_ws;                  // needs 65536*1536*4 = 384 MiB

  dim3 g1(65536 / 128, 1536 / 64);            // 512 x 24 blocks
  qkv_gemm_kernel<<<g1, 256, 0, stream>>>(X, W, bias, qkv);

  attn_softmax_kernel<<<2 * 8 * 8 * 64, 256, 0, stream>>>(qkv, out);
}